// torch_shift_72404558676308
// MI455X (gfx1250) — compile-verified
//
#include <hip/hip_runtime.h>
#include <stdint.h>

// torch_shift: out[c,d,hw] = x[c, d - s(c), hw], s(c) = (c/64) - 2, zero-filled.
// Shape (1, 320, 64, 64, 64) fp32. Pure HBM-bandwidth copy of 16KB D-slices.
// CDNA5 path: async global->LDS->global DMA (ASYNCcnt), no VGPR data movement.
// Both directions non-temporal: one-to-one slice mapping means zero reuse, so
// don't sweep the 192MB L2 with a 2x335MB stream.

#define HW_FLOATS 4096      // 64*64 floats per D-slice
#define DDIM 64
#define CDIM 320

typedef float v4f __attribute__((ext_vector_type(4)));   // native 16B vector

__global__ __launch_bounds__(256)
void torch_shift_kernel(const float* __restrict__ x, float* __restrict__ out) {
    __shared__ float smem[HW_FLOATS];   // 16 KB staging buffer

    const int b   = blockIdx.x;         // slice id: 0 .. C*D-1
    const int c   = b >> 6;             // channel 0..319
    const int d   = b & (DDIM - 1);     // depth slice 0..63
    const int s   = (c >> 6) - 2;       // per-group shift in [-2, 2]
    const int dsrc = d - s;             // source depth slice

    const int tid = threadIdx.x;        // 0..255
    float* dst = out + (long long)b * HW_FLOATS;

    if (dsrc >= 0 && dsrc < DDIM) {
        const float* src = x + (long long)(c * DDIM + dsrc) * HW_FLOATS;

        // Low 32 bits of the generic shared-memory pointer are the LDS byte
        // address (ISA 10.2 aperture rule).
        const unsigned lds_base = (unsigned)(unsigned long long)(&smem[0]);

        // 4 rounds x 256 lanes x 16B = 16 KB. Consecutive lanes -> consecutive
        // 16B, fully coalesced. Each wave loads/stores only its own LDS range,
        // so no workgroup barrier is needed between load and store phases.
#pragma unroll
        for (int j = 0; j < 4; ++j) {
            const unsigned gofs = (unsigned)((j * 256 + tid) * 16); // byte offset
            const unsigned lofs = lds_base + gofs;
            asm volatile("global_load_async_to_lds_b128 %0, %1, %2 th:TH_LOAD_NT"
                         :
                         : "v"(lofs), "v"(gofs), "s"(src)
                         : "memory");
        }
        asm volatile("s_wait_asynccnt 0" ::: "memory");
#pragma unroll
        for (int j = 0; j < 4; ++j) {
            const unsigned gofs = (unsigned)((j * 256 + tid) * 16);
            const unsigned lofs = lds_base + gofs;
            asm volatile("global_store_async_from_lds_b128 %0, %1, %2 th:TH_STORE_NT"
                         :
                         : "v"(gofs), "v"(lofs), "s"(dst)
                         : "memory");
        }
        // No trailing s_wait_asynccnt: S_ENDPGM implicitly waits for all
        // counters, and this block's LDS buffer is never reused.
    } else {
        // Zero-filled edge slice: streaming stores, bypass caches.
        const v4f z = (v4f)(0.0f);
        v4f* dst4 = (v4f*)dst;
#pragma unroll
        for (int j = 0; j < 4; ++j) {
            __builtin_nontemporal_store(z, dst4 + (j * 256 + tid));
        }
    }
}

extern "C" void kernel_launch(void* const* d_in, const int* in_sizes, int n_in,
                              void* d_out, int out_size, void* d_ws, size_t ws_size,
                              hipStream_t stream) {
    (void)in_sizes; (void)n_in; (void)d_ws; (void)ws_size;
    const float* x = (const float*)d_in[0];
    float* out = (float*)d_out;

    // out_size = 320*64*4096 = 83,886,080 floats -> 20480 slices of 4096 floats
    const int slices = out_size / HW_FLOATS;   // expected 20480
    torch_shift_kernel<<<dim3(slices), dim3(256), 0, stream>>>(x, out);
}